// SOM_26525718020321
// MI455X (gfx1250) — compile-verified
//
#include <hip/hip_runtime.h>
#include <math.h>

// Problem constants (fixed by the reference).
#define MG   128
#define NG   128
#define DIMD 256
#define BSZ  8192
#define KUNITS (MG * NG)   // 16384

typedef __attribute__((ext_vector_type(2))) float v2f;
typedef __attribute__((ext_vector_type(8))) float v8f;

// Exact f32 matrix op: D = A(16x4) * B(4x16) + C(16x16), wave32.
__device__ __forceinline__ v8f wmma_f32(v2f a, v2f b, v8f c) {
    return __builtin_amdgcn_wmma_f32_16x16x4_f32(
        /*neg_a=*/false, a, /*neg_b=*/false, b,
        /*c_mod=*/(short)0, c, /*reuse_a=*/false, /*reuse_b=*/false);
}

// ---------------------------------------------------------------------------
// Kernel 1: w_sq[k] = sum_d W[k,d]^2     (16384 threads)
// ---------------------------------------------------------------------------
__global__ void __launch_bounds__(256)
k_wsq(const float* __restrict__ W, float* __restrict__ wsq) {
    const int k = blockIdx.x * blockDim.x + threadIdx.x;
    const float4* w4 = (const float4*)(W + (size_t)k * DIMD);
    float acc = 0.0f;
#pragma unroll 8
    for (int i = 0; i < DIMD / 4; ++i) {
        float4 v = w4[i];
        acc += v.x * v.x + v.y * v.y + v.z * v.z + v.w * v.w;
    }
    wsq[k] = acc;
}

// ---------------------------------------------------------------------------
// Kernel 2: distance GEMM + argmin (BMU).
// One wave owns 16 data rows; A (16x256) stays resident in VGPRs; streams all
// 1024 column tiles of W doing 64 x V_WMMA_F32_16X16X4_F32 per tile.
// argmin uses d^2 - x_sq (monotone shift): val = wsq[col] - 2*dot.
// ---------------------------------------------------------------------------
__global__ void __launch_bounds__(256)
k_bmu(const float* __restrict__ X, const float* __restrict__ W,
      const float* __restrict__ wsq, int* __restrict__ bmu) {
    const int tid  = threadIdx.x;
    const int lane = tid & 31;
    const int wave = tid >> 5;
    const int l    = lane & 15;    // N index within tile / M index for A
    const int half = lane >> 4;    // selects K pair {0,1} vs {2,3}
    const int row0 = (blockIdx.x * 8 + wave) * 16;

    // A fragments: lane holds row (row0+l), K = 4s + 2*half + {0,1}
    v2f a[64];
    const float* arow = X + (size_t)(row0 + l) * DIMD + 2 * half;
#pragma unroll
    for (int s = 0; s < 64; ++s) {
        a[s].x = arow[4 * s + 0];
        a[s].y = arow[4 * s + 1];
    }

    float bestV[8];
    int   bestI[8];
#pragma unroll
    for (int r = 0; r < 8; ++r) { bestV[r] = 3.4e38f; bestI[r] = 0; }

    for (int tile = 0; tile < KUNITS / 16; ++tile) {
        const int col = tile * 16 + l;
        const float* brow = W + (size_t)col * DIMD + 2 * half;
        v8f c = {};
#pragma unroll
        for (int s = 0; s < 64; ++s) {
            v2f b;
            b.x = brow[4 * s + 0];
            b.y = brow[4 * s + 1];
            c = wmma_f32(a[s], b, c);
        }
        const float ws = wsq[col];
#pragma unroll
        for (int r = 0; r < 8; ++r) {
            float v = ws - 2.0f * c[r];   // lane's col, row = row0 + r + 8*half
            if (v < bestV[r] || (v == bestV[r] && col < bestI[r])) {
                bestV[r] = v;
                bestI[r] = col;
            }
        }
    }

    // Reduce across the 16 lanes of each half (XOR of bits 0..3 stays in-half).
#pragma unroll
    for (int r = 0; r < 8; ++r) {
        float v = bestV[r];
        int   i = bestI[r];
#pragma unroll
        for (int m = 8; m >= 1; m >>= 1) {
            float ov = __shfl_xor(v, m, 32);
            int   oi = __shfl_xor(i, m, 32);
            if (ov < v || (ov == v && oi < i)) { v = ov; i = oi; }
        }
        if (l == 0) bmu[row0 + r + 8 * half] = i;
    }
}

// ---------------------------------------------------------------------------
// Kernel 3: px[b,i], py[b,j] Gaussian factors (lr cancels in numer/denom).
// Schedule scalars computed on-device from `it` (it is a device pointer).
// ---------------------------------------------------------------------------
__global__ void __launch_bounds__(256)
k_pxy(const int* __restrict__ bmu, const int* __restrict__ itp,
      float* __restrict__ px, float* __restrict__ py) {
    const float START_SIGMA = 64.0f;               // max(M,N)/2
    const float TAU = 20.0f / logf(START_SIGMA);   // EPOCHS / ln(sigma0)
    const float sigma = START_SIGMA * expf(-(float)itp[0] / TAU);
    const float inv_d = 1.0f / (2.0f * sigma * sigma);

    const int b   = blockIdx.x * 2 + (threadIdx.x >> 7);
    const int idx = threadIdx.x & 127;
    const int k   = bmu[b];
    const float dr = (float)idx - (float)(k >> 7);    // row coord
    const float dc = (float)idx - (float)(k & 127);   // col coord
    px[(size_t)b * MG + idx] = expf(-dr * dr * inv_d);
    py[(size_t)b * NG + idx] = expf(-dc * dc * inv_d);
}

// ---------------------------------------------------------------------------
// Kernel 4: denom = Px^T * Py  ([128,128], reduce over 8192) via WMMA f32.
// 64 tiles of 16x16, one wave per tile.
// ---------------------------------------------------------------------------
__global__ void __launch_bounds__(256)
k_denom(const float* __restrict__ px, const float* __restrict__ py,
        float* __restrict__ denom) {
    const int tid  = threadIdx.x;
    const int lane = tid & 31;
    const int wave = tid >> 5;
    const int l    = lane & 15;
    const int half = lane >> 4;
    const int t     = blockIdx.x * 8 + wave;     // 0..63
    const int iBase = (t >> 3) * 16;
    const int jBase = (t & 7) * 16;

    v8f c = {};
    for (int b = 0; b < BSZ; b += 4) {
        const int bb = b + 2 * half;
        v2f a, bf;
        a.x  = px[(size_t)bb * MG + iBase + l];
        a.y  = px[(size_t)(bb + 1) * MG + iBase + l];
        bf.x = py[(size_t)bb * NG + jBase + l];
        bf.y = py[(size_t)(bb + 1) * NG + jBase + l];
        c = wmma_f32(a, bf, c);
    }
#pragma unroll
    for (int r = 0; r < 8; ++r)
        denom[(size_t)(iBase + r + 8 * half) * NG + jBase + l] = c[r];
}

// ---------------------------------------------------------------------------
// Kernel 5: numer = neighbor^T * data  ([16384,256], reduce over 8192) and
// final output. A fragments are built on the fly as px*py (2 VALU mults),
// each reused across 4 d-tiles (64 output dims per wave).
// ---------------------------------------------------------------------------
__global__ void __launch_bounds__(256)
k_update(const float* __restrict__ X, const float* __restrict__ px,
         const float* __restrict__ py, const float* __restrict__ denom,
         const float* __restrict__ W, float* __restrict__ out) {
    const int tid  = threadIdx.x;
    const int lane = tid & 31;
    const int wave = tid >> 5;
    const int l    = lane & 15;
    const int half = lane >> 4;

    const int gw    = blockIdx.x * 8 + wave;   // 0..4095
    const int kTile = gw >> 2;                 // 0..1023 (16 units each)
    const int dBase = (gw & 3) * 64;           // 4 x 16 dims per wave
    const int kBase = kTile * 16;
    const int i     = kBase >> 7;              // SOM row (fixed within tile)
    const int j0    = kBase & 127;             // SOM col base

    v8f c0 = {}, c1 = {}, c2 = {}, c3 = {};
    for (int b = 0; b < BSZ; b += 4) {
        const int bb = b + 2 * half;
        const float p0 = px[(size_t)bb * MG + i];
        const float p1 = px[(size_t)(bb + 1) * MG + i];
        v2f a;
        a.x = p0 * py[(size_t)bb * NG + j0 + l];
        a.y = p1 * py[(size_t)(bb + 1) * NG + j0 + l];

        const float* x0 = X + (size_t)bb * DIMD + dBase + l;
        const float* x1 = X + (size_t)(bb + 1) * DIMD + dBase + l;
        v2f bf;
        bf.x = x0[0];  bf.y = x1[0];  c0 = wmma_f32(a, bf, c0);
        bf.x = x0[16]; bf.y = x1[16]; c1 = wmma_f32(a, bf, c1);
        bf.x = x0[32]; bf.y = x1[32]; c2 = wmma_f32(a, bf, c2);
        bf.x = x0[48]; bf.y = x1[48]; c3 = wmma_f32(a, bf, c3);
    }

    float den[8];
#pragma unroll
    for (int r = 0; r < 8; ++r)
        den[r] = denom[kBase + r + 8 * half];

    const v8f* cs[4] = { &c0, &c1, &c2, &c3 };
#pragma unroll
    for (int dt = 0; dt < 4; ++dt) {
#pragma unroll
        for (int r = 0; r < 8; ++r) {
            const int k = kBase + r + 8 * half;
            const int d = dBase + dt * 16 + l;
            const float v = (*cs[dt])[r];
            out[(size_t)k * DIMD + d] =
                (den[r] != 0.0f) ? (v / den[r]) : W[(size_t)k * DIMD + d];
        }
    }
}

// ---------------------------------------------------------------------------
// Launch: inputs are {data[8192,256] f32, all_weights[128,128,256] f32, it i32}
// ---------------------------------------------------------------------------
extern "C" void kernel_launch(void* const* d_in, const int* in_sizes, int n_in,
                              void* d_out, int out_size, void* d_ws, size_t ws_size,
                              hipStream_t stream) {
    const float* X  = (const float*)d_in[0];
    const float* W  = (const float*)d_in[1];
    const int*   it = (const int*)d_in[2];
    float* out = (float*)d_out;

    // Workspace layout (floats): ~8.6 MB total.
    float* ws    = (float*)d_ws;
    float* wsq   = ws;                               // 16384
    int*   bmu   = (int*)(ws + KUNITS);              // 8192
    float* px    = ws + KUNITS + BSZ;                // 8192*128
    float* py    = px + (size_t)BSZ * MG;            // 8192*128
    float* denom = py + (size_t)BSZ * NG;            // 16384

    k_wsq   <<<KUNITS / 256, 256, 0, stream>>>(W, wsq);
    k_bmu   <<<BSZ / (8 * 16), 256, 0, stream>>>(X, W, wsq, bmu);   // 64 WGs
    k_pxy   <<<BSZ / 2, 256, 0, stream>>>(bmu, it, px, py);
    k_denom <<<8, 256, 0, stream>>>(px, py, denom);                 // 64 waves
    k_update<<<(KUNITS / 16) * 4 / 8, 256, 0, stream>>>(X, px, py, denom, W, out); // 512 WGs
}